// Wassertein_35656818492192
// MI455X (gfx1250) — compile-verified
//
#include <hip/hip_runtime.h>

// CDNA5 / gfx1250 kernel for the Dempster-Shafer "Wasserstein-to-focal" map.
// Per pixel: out[s] = (bel_s + mOmega/2 - c_s)^2 + (mOmega/2)^2/3 for s=1..6,
//            out[0] = out[7] = 0.   (c_s = 1 for odd s; FOCAL = 1)
//
// The linear part is a per-pixel [4] x [4x8] matmul -> V_WMMA_F32_16X16X4_F32:
// 16 pixels form the M dimension, K = 4 input channels; the constant
// subtraction (-sum_x) is folded into the C accumulator, the +mOmega/2 term is
// folded into the Omega row of B, and B column 8 = [0,0,0,1] so D[.,8] carries
// mOmega out of the same WMMA for the quadratic "dif" term.
//
// Epilogue: raw D is stored to LDS unconditionally (16 pix x 16 cols / wave,
// lowered to ds_store_2addr_b32 pairs), then read back transposed (b128) so
// the square / +q / channel-mask work happens in the coalesced output layout
// with zero wave divergence.
//
// Loop control is forced wave-uniform via readfirstlane so the hot loop is
// SALU-controlled (no EXEC masking around the WMMA, as the ISA requires).
//
// Memory-bound: ~92 MB traffic @ 23.3 TB/s => ~4 us floor; all global traffic
// is coalesced b64 loads / b128 stores.

typedef __attribute__((ext_vector_type(2))) float v2f;
typedef __attribute__((ext_vector_type(4))) float v4f;
typedef __attribute__((ext_vector_type(8))) float v8f;

#define WAVES_PER_BLOCK 8
#define BLOCK_THREADS   (WAVES_PER_BLOCK * 32)

__global__ __launch_bounds__(BLOCK_THREADS)
void dst_focal_wmma_kernel(const float* __restrict__ in,
                           float* __restrict__ out,
                           int ntiles,          // number of 16-pixel tiles
                           int npix,            // total pixels (tail handling)
                           int nwaves_total) {
  __shared__ float lds[WAVES_PER_BLOCK * 256];   // 1 KB per wave: 16 pix x 16 cols

  const int tid  = threadIdx.x;
  const int lane = tid & 31;
  const int wv   = tid >> 5;
  const int n  = lane & 15;     // column index (channel s for n<8)
  const int hf = lane >> 4;     // half-wave select

  // Wave-uniform (scalar) loop parameters: keeps loop control in SALU and
  // EXEC all-1s through the WMMA.
  const int gwave  = __builtin_amdgcn_readfirstlane(blockIdx.x * WAVES_PER_BLOCK + wv);
  const int nt     = __builtin_amdgcn_readfirstlane(ntiles);
  const int stride = __builtin_amdgcn_readfirstlane(nwaves_total);

  // ---- B matrix (4x16 f32): lanes 0-15 hold rows K=0 (v0) / K=1 (v1),
  //      lanes 16-31 hold rows K=2 (v0) / K=3 (v1).
  //  cols 1..6 : membership bits for k<3, 0.5 for k=3 (folds +mOmega/2 in)
  //  col 8     : [0,0,0,1]  -> D[.,8] = mOmega
  //  others    : 0
  const int k0 = hf * 2;
  float bv[2];
  #pragma unroll
  for (int i = 0; i < 2; ++i) {
    const int k = k0 + i;
    float val = 0.0f;
    if (n >= 1 && n <= 6)  val = (k < 3) ? (float)((n >> k) & 1) : 0.5f;
    else if (n == 8)       val = (k == 3) ? 1.0f : 0.0f;
    bv[i] = val;
  }
  v2f Bm; Bm.x = bv[0]; Bm.y = bv[1];

  // ---- C matrix: -1 in columns 1,3,5 (folds -sum_x in); same for all rows.
  const float cval = ((n == 1) || (n == 3) || (n == 5)) ? -1.0f : 0.0f;
  v8f Cm;
  #pragma unroll
  for (int j = 0; j < 8; ++j) Cm[j] = cval;

  // ---- LDS addressing (floats, per-wave 256-float slice) ----
  float* wlds = lds + wv * 256;
  // Store: lane writes D[j] at (pix = j + hf*8, col = n) -> j*16 + hf*128 + n.
  const int stbase = hf * 128 + n;
  // Read: lane owns pixel p = lane>>1, channels 4c..4c+3 with c = lane&1.
  const int p  = lane >> 1;
  const int c  = lane & 1;
  const int rdbase = p * 16 + c * 4;
  const int mobase = p * 16 + 8;          // mOmega column
  // Channel mask: zero out channel 0 (c=0,elem0) and channel 7 (c=1,elem3).
  v4f cm;
  cm.x = c ? 1.0f : 0.0f;
  cm.y = 1.0f;
  cm.z = 1.0f;
  cm.w = c ? 0.0f : 1.0f;

  const v2f* in2 = (const v2f*)in;

  // Scalar-controlled grid-stride loop over 16-pixel tiles.
  for (int tile = gwave; tile < nt; tile += stride) {
    const int p0 = tile << 4;

    // A matrix load (ISA 16x4 f32 A layout), one coalesced b64 per lane:
    // lanes 0-15: pixel p0+n channels {0,1}; lanes 16-31: pixel p0+n {2,3}.
    v2f Am = in2[(size_t)(p0 + n) * 2 + hf];

    // Prefetch next tile of this wave (scalar guard -> s_cbranch).
    {
      const int tn = tile + stride;
      if (tn < nt)
        __builtin_prefetch(&in2[(size_t)((tn << 4) + n) * 2 + hf], 0, 3);
    }

    // D[m, s] = sum_k m_k * B[k,s] + C[m,s]
    v8f D = __builtin_amdgcn_wmma_f32_16x16x4_f32(
        /*neg_a=*/false, Am, /*neg_b=*/false, Bm,
        /*c_mod=*/(short)0, Cm, /*reuse_a=*/false, /*reuse_b=*/false);

    // Unconditional raw-D transpose store: one base, immediate offsets j*64B.
    #pragma unroll
    for (int j = 0; j < 8; ++j)
      wlds[j * 16 + stbase] = D[j];

    // Read back in output layout + mOmega, finish the quadratic per element.
    v4f dv = *(const v4f*)(wlds + rdbase);
    const float mo = wlds[mobase];
    const float q  = mo * mo * (1.0f / 12.0f);
    v4f o;
    o.x = (dv.x * dv.x + q) * cm.x;
    o.y = (dv.y * dv.y + q) * cm.y;
    o.z = (dv.z * dv.z + q) * cm.z;
    o.w = (dv.w * dv.w + q) * cm.w;
    *(v4f*)(out + (size_t)p0 * 8 + lane * 4) = o;   // coalesced b128
  }

  // ---- Scalar tail for npix % 16 != 0 (not hit for the given shape).
  const int tail0 = ntiles << 4;
  for (int pp = tail0 + blockIdx.x * BLOCK_THREADS + tid; pp < npix;
       pp += gridDim.x * BLOCK_THREADS) {
    const v4f m = *(const v4f*)(in + (size_t)pp * 4);
    const float mw = m.w;
    const float q  = mw * mw * (1.0f / 12.0f);
    #pragma unroll
    for (int s = 0; s < 8; ++s) {
      float r = 0.0f;
      if (s >= 1 && s <= 6) {
        float bel = 0.0f;
        if (s & 1) bel += m.x;
        if (s & 2) bel += m.y;
        if (s & 4) bel += m.z;
        const float t = bel + 0.5f * mw - (float)(s & 1);
        r = t * t + q;
      }
      out[(size_t)pp * 8 + s] = r;
    }
  }
}

extern "C" void kernel_launch(void* const* d_in, const int* in_sizes, int n_in,
                              void* d_out, int out_size, void* d_ws, size_t ws_size,
                              hipStream_t stream) {
  const float* in = (const float*)d_in[0];
  float* out = (float*)d_out;

  const int npix   = in_sizes[0] / 4;   // 4 floats per pixel
  const int ntiles = npix / 16;         // 16 pixels per WMMA tile

  // ~8 tiles per wave; memory-bound, so we just need enough waves in flight
  // to cover HBM latency.
  int nwaves = (ntiles + 7) / 8;
  int nblocks = (nwaves + WAVES_PER_BLOCK - 1) / WAVES_PER_BLOCK;
  if (nblocks < 1) nblocks = 1;
  const int nwaves_total = nblocks * WAVES_PER_BLOCK;

  dst_focal_wmma_kernel<<<nblocks, BLOCK_THREADS, 0, stream>>>(
      in, out, ntiles, npix, nwaves_total);
}